// MaxEntIRLLoss_78993038508419
// MI455X (gfx1250) — compile-verified
//
#include <hip/hip_runtime.h>
#include <hip/hip_bf16.h>

#define B_   64
#define T_   256
#define H_   512
#define W_   1024
#define HW_  (H_ * W_)
#define EPSF 1e-05f
#define CHUNKS 64          // blocks per batch in reduction
#define RBLOCK 256         // threads per reduce block (8 waves)

typedef __attribute__((ext_vector_type(2))) float v2f;
typedef __attribute__((ext_vector_type(8))) float v8f;

// ---------------------------------------------------------------------------
// Wave32 sum via V_WMMA_F32_16X16X4_F32:
//   A (16x4, f32): lane L<16 -> A[L][0]=p, lane L>=16 -> A[L-16][2]=p (other K=0)
//   B = all ones  => D[m][n] = p_m + p_{m+16} for every n
//   Per-lane sum of the 8 D VGPRs = half-total (rows 0..7 for lanes<16,
//   rows 8..15 for lanes>=16); one xor-16 shuffle completes the wave sum.
// EXEC must be all-ones at the call site.
// ---------------------------------------------------------------------------
__device__ __forceinline__ float wave_sum_wmma(float p) {
    v2f a; a.x = p;    a.y = 0.0f;
    v2f b; b.x = 1.0f; b.y = 1.0f;
    v8f c = {};
    c = __builtin_amdgcn_wmma_f32_16x16x4_f32(
            /*neg_a=*/false, a, /*neg_b=*/false, b,
            /*c_mod=*/(short)0, c, /*reuse_a=*/false, /*reuse_b=*/false);
    float s = c[0] + c[1] + c[2] + c[3] + c[4] + c[5] + c[6] + c[7];
    s += __shfl_xor(s, 16, 32);
    return s;
}

// ---------------------------------------------------------------------------
// K0: zero the svf region of d_out (scalar b32 stores: region is only 4B aligned)
// ---------------------------------------------------------------------------
__global__ void k_zero(float* __restrict__ svf, long long n) {
    long long i = (long long)blockIdx.x * blockDim.x + threadIdx.x;
    long long stride = (long long)gridDim.x * blockDim.x;
    for (; i < n; i += stride) svf[i] = 0.0f;
}

// ---------------------------------------------------------------------------
// K1: interpolate trajectory points, write interp_pts output, scatter 1.0f
//     into the binary visitation map (benign races: all writers store 1.0f).
// ---------------------------------------------------------------------------
__global__ void k_interp_scatter(const float* __restrict__ gt,
                                 float* __restrict__ svf,
                                 float* __restrict__ interp,
                                 int N, int max_steps) {
    int idx = blockIdx.x * blockDim.x + threadIdx.x;
    int total = B_ * N;
    if (idx >= total) return;
    int b = idx / N;
    int p = idx - b * N;

    float px, py;
    int seg_total = (T_ - 1) * max_steps;
    if (p < seg_total) {
        int s = p / max_steps;
        int k = p - s * max_steps;
        float denom = (max_steps > 1) ? (float)(max_steps - 1) : 1.0f;
        float t = (float)k / denom;
        const float* g0 = gt + (long long)(b * T_ + s) * 9;
        float x0 = g0[2] * 0.5f, y0 = g0[5] * 0.5f;   // gt[b,s,0,2]/2, gt[b,s,1,2]/2
        float x1 = g0[11] * 0.5f, y1 = g0[14] * 0.5f; // next t index (+9)
        px = x0 + t * (x1 - x0);
        py = y0 + t * (y1 - y0);
    } else {
        const float* g = gt + (long long)(b * T_ + (T_ - 1)) * 9;
        px = g[2] * 0.5f;
        py = g[5] * 0.5f;
    }

    interp[(long long)idx * 2 + 0] = px;
    interp[(long long)idx * 2 + 1] = py;

    float cx = fminf(fmaxf(px, 0.0f), (float)(H_ - 1));
    float cy = fminf(fmaxf(py, 0.0f), (float)(W_ - 1));
    int xi = (int)cx;          // non-negative -> truncation == floor == int32 cast
    int yi = (int)cy;
    svf[(long long)b * HW_ + (long long)xi * W_ + yi] = 1.0f;
}

// ---------------------------------------------------------------------------
// K2: per-(batch, chunk) reduction of 4 quantities:
//     cnt = sum(svf_bin), sr = sum(svf_bin * reward),
//     es  = sum(exp_svf), er = sum(exp_svf * reward)
// Wave sum via WMMA, then fixed-order cross-wave sum -> partials[b][chunk][4].
// ---------------------------------------------------------------------------
__global__ void k_reduce(const float* __restrict__ svf,
                         const float* __restrict__ exp_svf,
                         const float* __restrict__ reward,
                         float* __restrict__ partials,
                         int per_chunk) {
    int b = blockIdx.y;
    int chunk = blockIdx.x;
    long long base = (long long)b * HW_ + (long long)chunk * per_chunk;

    float cnt = 0.0f, sr = 0.0f, es = 0.0f, er = 0.0f;
    for (int i = threadIdx.x; i < per_chunk; i += blockDim.x) {
        float sv = svf[base + i];
        float rw = reward[base + i];
        float ex = exp_svf[base + i];
        cnt += sv;
        sr  += sv * rw;
        es  += ex;
        er  += ex * rw;
    }

    // per_chunk is a multiple of blockDim -> no divergence; EXEC all-ones here
    cnt = wave_sum_wmma(cnt);
    sr  = wave_sum_wmma(sr);
    es  = wave_sum_wmma(es);
    er  = wave_sum_wmma(er);

    __shared__ float ls[RBLOCK / 32][4];
    int wave = threadIdx.x >> 5;
    int lane = threadIdx.x & 31;
    if (lane == 0) { ls[wave][0] = cnt; ls[wave][1] = sr; ls[wave][2] = es; ls[wave][3] = er; }
    __syncthreads();
    if (threadIdx.x == 0) {
        float a0 = 0.0f, a1 = 0.0f, a2 = 0.0f, a3 = 0.0f;
        for (int w = 0; w < RBLOCK / 32; ++w) {
            a0 += ls[w][0]; a1 += ls[w][1]; a2 += ls[w][2]; a3 += ls[w][3];
        }
        float* pp = partials + ((long long)b * gridDim.x + chunk) * 4;
        pp[0] = a0; pp[1] = a1; pp[2] = a2; pp[3] = a3;
    }
}

// ---------------------------------------------------------------------------
// K3: finalize per-batch stats (fixed-order sums), write inv_cnt + loss
// ---------------------------------------------------------------------------
__global__ void k_finalize(const float* __restrict__ partials,
                           float* __restrict__ inv_cnt,
                           float* __restrict__ loss_out,
                           int chunks) {
    __shared__ float diff[B_];
    int b = threadIdx.x;   // 0..63
    float cnt = 0.0f, sr = 0.0f, es = 0.0f, er = 0.0f;
    for (int c = 0; c < chunks; ++c) {
        const float* pp = partials + ((long long)b * chunks + c) * 4;
        cnt += pp[0]; sr += pp[1]; es += pp[2]; er += pp[3];
    }
    float ic = 1.0f / (cnt + EPSF);
    inv_cnt[b] = ic;
    float svf_r = sr * ic;
    float exp_r = er / (es + EPSF);
    diff[b] = exp_r - svf_r;
    __syncthreads();
    if (b == 0) {
        float s = 0.0f;
        for (int i = 0; i < B_; ++i) s += diff[i];
        loss_out[0] = s * (1.0f / (float)B_);   // MAXENT_WEIGHT == 1.0
    }
}

// ---------------------------------------------------------------------------
// K4: normalize svf in place: binary * inv_cnt[b]
// ---------------------------------------------------------------------------
__global__ void k_normalize(float* __restrict__ svf,
                            const float* __restrict__ inv_cnt,
                            long long n) {
    long long i = (long long)blockIdx.x * blockDim.x + threadIdx.x;
    long long stride = (long long)gridDim.x * blockDim.x;
    for (; i < n; i += stride) {
        int b = (int)(i / HW_);
        svf[i] *= inv_cnt[b];
    }
}

extern "C" void kernel_launch(void* const* d_in, const int* in_sizes, int n_in,
                              void* d_out, int out_size, void* d_ws, size_t ws_size,
                              hipStream_t stream) {
    const float* gt      = (const float*)d_in[0];   // (B,T,3,3)
    const float* exp_svf = (const float*)d_in[1];   // (B,H,W)
    const float* reward  = (const float*)d_in[2];   // (B,H,W)

    float* out       = (float*)d_out;
    float* loss_out  = out;                         // [0]
    float* svf_out   = out + 1;                     // [1 .. 1+B*H*W)
    float* interp    = out + 1 + (long long)B_ * HW_;

    // Recover max_steps from the (fixed) output size:
    //   out_size = 1 + B*H*W + B*((T-1)*ms + 1)*2
    long long interp_elems = (long long)out_size - 1 - (long long)B_ * HW_;
    int N = (int)(interp_elems / (2LL * B_));
    if (N < 1) N = 1;
    int max_steps = (N - 1) / (T_ - 1);
    if (max_steps < 1) max_steps = 1;

    // Workspace layout
    float* partials = (float*)d_ws;                       // B*CHUNKS*4 floats
    float* inv_cnt  = partials + (long long)B_ * CHUNKS * 4; // B floats

    const long long n_svf = (long long)B_ * HW_;

    // K0: zero visitation map
    {
        int blocks = 8192;  // grid-stride
        k_zero<<<blocks, 256, 0, stream>>>(svf_out, n_svf);
    }
    // K1: interpolate + scatter
    {
        int total = B_ * N;
        int blocks = (total + 255) / 256;
        k_interp_scatter<<<blocks, 256, 0, stream>>>(gt, svf_out, interp, N, max_steps);
    }
    // K2: reductions (WMMA wave sums)
    {
        dim3 grid(CHUNKS, B_);
        int per_chunk = HW_ / CHUNKS;   // 8192, multiple of RBLOCK
        k_reduce<<<grid, RBLOCK, 0, stream>>>(svf_out, exp_svf, reward, partials, per_chunk);
    }
    // K3: finalize stats + loss
    k_finalize<<<1, B_, 0, stream>>>(partials, inv_cnt, loss_out, CHUNKS);
    // K4: normalize svf in place
    {
        int blocks = 8192;
        k_normalize<<<blocks, 256, 0, stream>>>(svf_out, inv_cnt, n_svf);
    }
}